// GINTopK_37065567765121
// MI455X (gfx1250) — compile-verified
//
#include <hip/hip_runtime.h>
#include <math.h>

typedef float v2f __attribute__((ext_vector_type(2)));
typedef float v8f __attribute__((ext_vector_type(8)));

#define GG 64   // graphs

// ---------------- utility kernels ----------------

__global__ void fill_f_kernel(float* __restrict__ p, float v, int n) {
    int i = blockIdx.x * blockDim.x + threadIdx.x;
    if (i < n) p[i] = v;
}

__global__ void fill_i_kernel(int* __restrict__ p, int v, int n) {
    int i = blockIdx.x * blockDim.x + threadIdx.x;
    if (i < n) p[i] = v;
}

__global__ void copy_f4_kernel(float4* __restrict__ dst, const float4* __restrict__ src, int n4) {
    int i = blockIdx.x * blockDim.x + threadIdx.x;
    if (i < n4) dst[i] = src[i];
}

// starts[g] = lower_bound(batch, g) for g in [0,64]; batch is sorted.
__global__ void starts_kernel(const int* __restrict__ batch, int* __restrict__ starts, int N) {
    int g = threadIdx.x;
    if (g > GG) return;
    int lo = 0, hi = N;
    while (lo < hi) {
        int m = (lo + hi) >> 1;
        if (batch[m] < g) lo = m + 1; else hi = m;
    }
    starts[g] = lo;
}

// ---------------- edge aggregation (scatter-add) ----------------
// agg[dst][d] += x[src][d] for every edge; masking is redundant because
// dead-node features are exactly zero and dead destinations get re-masked.
__global__ __launch_bounds__(256) void scatter_agg_kernel(
        const int* __restrict__ esrc, const int* __restrict__ edst,
        const float* __restrict__ x, float* __restrict__ agg, int E) {
    int t = blockIdx.x * 256 + threadIdx.x;
    int e = t >> 6;
    int d = t & 63;
    if (e >= E) return;
    int s  = esrc[e];
    int dd = edst[e];
    float v = x[(size_t)s * 64 + d];
    if (v != 0.0f) atomicAdd(&agg[(size_t)dd * 64 + d], v);
}

// ---------------- GIN MLP via V_WMMA_F32_16X16X4_F32 ----------------
// One wave per 16-node tile. h = x + agg;  out = mask * relu(relu(h@W1+b1)@W2+b2)
// Written in-place into x (each wave stages its rows into LDS first).
__global__ __launch_bounds__(128) void gin_mlp_kernel(
        const float* __restrict__ x, const float* __restrict__ agg,
        const float* __restrict__ W1, const float* __restrict__ b1,
        const float* __restrict__ W2, const float* __restrict__ b2,
        const int* __restrict__ mask, float* __restrict__ xout, int n) {
    __shared__ float lds_in[4][16 * 64];
    __shared__ float lds_h [4][16 * 64];
    const int wave = threadIdx.x >> 5;
    const int lane = threadIdx.x & 31;
    const int m0 = (blockIdx.x * 4 + wave) * 16;
    if (m0 >= n) return;

    float* s_in = lds_in[wave];
    float* s_h  = lds_h [wave];

    // Stage 16x64 input tile (h = x + agg) into LDS, coalesced.
    #pragma unroll
    for (int i = 0; i < 32; ++i) {
        int idx = i * 32 + lane;
        s_in[idx] = x[(size_t)m0 * 64 + idx] + agg[(size_t)m0 * 64 + idx];
    }
    asm volatile("s_wait_dscnt 0" ::: "memory");  // per-wave LDS RAW fence

    const int half = lane >> 4;   // 0: lanes 0-15, 1: lanes 16-31
    const int r16  = lane & 15;

    // ---- Layer 1: s_h = relu(s_in @ W1 + b1) ----
    for (int nb = 0; nb < 4; ++nb) {
        const int col = nb * 16 + r16;
        const float bias = b1[col];
        v8f c = {bias, bias, bias, bias, bias, bias, bias, bias};
        #pragma unroll
        for (int k0 = 0; k0 < 64; k0 += 4) {
            v2f va, vb;
            va.x = s_in[r16 * 64 + k0 + 2 * half + 0];
            va.y = s_in[r16 * 64 + k0 + 2 * half + 1];
            vb.x = W1[(k0 + 2 * half + 0) * 64 + col];
            vb.y = W1[(k0 + 2 * half + 1) * 64 + col];
            c = __builtin_amdgcn_wmma_f32_16x16x4_f32(
                    false, va, false, vb, (short)0, c, false, false);
        }
        #pragma unroll
        for (int r = 0; r < 8; ++r) {
            int row = r + half * 8;
            float v = c[r] > 0.0f ? c[r] : 0.0f;
            s_h[row * 64 + col] = v;
        }
    }
    asm volatile("s_wait_dscnt 0" ::: "memory");

    // ---- Layer 2: xout = mask * relu(s_h @ W2 + b2) ----
    for (int nb = 0; nb < 4; ++nb) {
        const int col = nb * 16 + r16;
        const float bias = b2[col];
        v8f c = {bias, bias, bias, bias, bias, bias, bias, bias};
        #pragma unroll
        for (int k0 = 0; k0 < 64; k0 += 4) {
            v2f va, vb;
            va.x = s_h[r16 * 64 + k0 + 2 * half + 0];
            va.y = s_h[r16 * 64 + k0 + 2 * half + 1];
            vb.x = W2[(k0 + 2 * half + 0) * 64 + col];
            vb.y = W2[(k0 + 2 * half + 1) * 64 + col];
            c = __builtin_amdgcn_wmma_f32_16x16x4_f32(
                    false, va, false, vb, (short)0, c, false, false);
        }
        #pragma unroll
        for (int r = 0; r < 8; ++r) {
            int row  = r + half * 8;
            int node = m0 + row;
            if (node < n) {
                float v = c[r] > 0.0f ? c[r] : 0.0f;
                xout[(size_t)node * 64 + col] = mask[node] ? v : 0.0f;
            }
        }
    }
}

// ---------------- scores + order-preserving keys ----------------
__device__ __forceinline__ unsigned fkey(float f) {
    unsigned u = __float_as_uint(f);
    return (u & 0x80000000u) ? ~u : (u | 0x80000000u);
}

__global__ __launch_bounds__(256) void score_kernel(
        const float* __restrict__ x, const float* __restrict__ w,
        const int* __restrict__ mask, float* __restrict__ score,
        unsigned* __restrict__ key, int N) {
    __shared__ float s_w[64];
    __shared__ float s_norm;
    int tid = threadIdx.x;
    if (tid < 64) s_w[tid] = w[tid];
    __syncthreads();
    if (tid == 0) {
        float s = 0.0f;
        for (int j = 0; j < 64; ++j) s += s_w[j] * s_w[j];
        s_norm = sqrtf(s);
    }
    __syncthreads();
    int i = blockIdx.x * 256 + tid;
    if (i < N) {
        const float* xr = x + (size_t)i * 64;
        float acc = 0.0f;
        #pragma unroll
        for (int j = 0; j < 64; ++j) acc += xr[j] * s_w[j];
        float sc = acc / s_norm;
        score[i] = sc;
        key[i] = mask[i] ? fkey(sc) : 0u;  // dead nodes: key 0 (< any finite key)
    }
}

// ---------------- exact per-graph TopK (block per graph) ----------------
__device__ int block_reduce_sum_i(int v, int* sred) {
    int tid = threadIdx.x;
    __syncthreads();
    sred[tid] = v;
    __syncthreads();
    for (int s = 128; s > 0; s >>= 1) {
        if (tid < s) sred[tid] += sred[tid + s];
        __syncthreads();
    }
    return sred[0];
}

__global__ __launch_bounds__(256) void topk_kernel(
        const unsigned* __restrict__ key, const int* __restrict__ starts,
        int* __restrict__ mask) {
    __shared__ int sred[256];
    int g = blockIdx.x, tid = threadIdx.x;
    int beg = starts[g], end = starts[g + 1];

    int c = 0;
    for (int i = beg + tid; i < end; i += 256) c += (key[i] != 0u);
    int n_alive = block_reduce_sum_i(c, sred);
    int k = (n_alive > 0) ? (int)ceilf(0.8f * (float)n_alive) : 0;  // match JAX f32

    unsigned T = 0u;
    if (k > 0) {
        // max T with count(alive key >= T) >= k  (bit-wise binary search)
        for (int bit = 31; bit >= 0; --bit) {
            unsigned cand = T | (1u << bit);
            int cc = 0;
            for (int i = beg + tid; i < end; i += 256) cc += (key[i] >= cand);
            int tot = block_reduce_sum_i(cc, sred);
            if (tot >= k) T = cand;
        }
    }
    int cg = 0;
    if (k > 0) for (int i = beg + tid; i < end; i += 256) cg += (key[i] > T);
    int c_gt = block_reduce_sum_i(cg, sred);

    for (int i = beg + tid; i < end; i += 256)
        mask[i] = (k > 0 && key[i] > T) ? 1 : 0;
    __syncthreads();
    if (tid == 0 && k > 0) {
        int r = k - c_gt;                     // stable index-order ties (lexsort)
        for (int i = beg; i < end && r > 0; ++i)
            if (key[i] == T) { mask[i] = 1; --r; }
    }
}

// ---------------- gating: x = keep ? x*tanh(score) : 0 ----------------
__global__ __launch_bounds__(256) void gate_kernel(
        float* __restrict__ x, const float* __restrict__ score,
        const int* __restrict__ mask, int total) {
    int i = blockIdx.x * 256 + threadIdx.x;
    if (i >= total) return;
    int node = i >> 6;
    x[i] = mask[node] ? x[i] * tanhf(score[node]) : 0.0f;
}

// ---------------- readout accumulate: r += [max, mean] ----------------
__global__ __launch_bounds__(256) void readout_kernel(
        const float* __restrict__ x, const int* __restrict__ mask,
        const int* __restrict__ starts, float* __restrict__ r) {
    __shared__ float smax[256], ssum[256];
    __shared__ int scnt[4];
    int g = blockIdx.x, tid = threadIdx.x;
    int grp = tid >> 6, d = tid & 63;
    int beg = starts[g], end = starts[g + 1];
    float mx = -INFINITY, sm = 0.0f;
    int c = 0;
    for (int i = beg + grp; i < end; i += 4) {
        if (mask[i]) {
            float v = x[(size_t)i * 64 + d];
            mx = fmaxf(mx, v);
            sm += v;
            c++;
        }
    }
    smax[tid] = mx; ssum[tid] = sm;
    if (d == 0) scnt[grp] = c;
    __syncthreads();
    if (grp == 0) {
        float M = mx, S = sm;
        for (int j = 1; j < 4; ++j) {
            M = fmaxf(M, smax[j * 64 + d]);
            S += ssum[j * 64 + d];
        }
        int C = scnt[0] + scnt[1] + scnt[2] + scnt[3];
        r[g * 128 + d]      += M;
        r[g * 128 + 64 + d] += S / fmaxf((float)C, 1.0f);
    }
}

// ---------------- final 3-layer head (64 graphs) ----------------
__global__ __launch_bounds__(64) void final_mlp_kernel(
        const float* __restrict__ r,
        const float* __restrict__ W1, const float* __restrict__ b1,
        const float* __restrict__ W2, const float* __restrict__ b2,
        const float* __restrict__ W3, const float* __restrict__ b3,
        float* __restrict__ out) {
    int g = threadIdx.x;
    if (g >= GG) return;
    float h1[64], h2[32];
    for (int o = 0; o < 64; ++o) {
        float s = b1[o];
        for (int i = 0; i < 128; ++i) s += r[g * 128 + i] * W1[i * 64 + o];
        h1[o] = s > 0.0f ? s : 0.0f;
    }
    for (int o = 0; o < 32; ++o) {
        float s = b2[o];
        for (int i = 0; i < 64; ++i) s += h1[i] * W2[i * 32 + o];
        h2[o] = s > 0.0f ? s : 0.0f;
    }
    for (int o = 0; o < 2; ++o) {
        float s = b3[o];
        for (int i = 0; i < 32; ++i) s += h2[i] * W3[i * 2 + o];
        out[g * 2 + o] = s;
    }
}

// ---------------- host orchestration ----------------
extern "C" void kernel_launch(void* const* d_in, const int* in_sizes, int n_in,
                              void* d_out, int out_size, void* d_ws, size_t ws_size,
                              hipStream_t stream) {
    (void)n_in; (void)out_size; (void)ws_size;
    const float* x_in  = (const float*)d_in[0];
    const int*   esrc  = (const int*)d_in[1];
    const int*   edst  = (const int*)d_in[2];
    const int*   batch = (const int*)d_in[3];
    const int N = in_sizes[3];
    const int E = in_sizes[1];

    // workspace layout
    char* w = (char*)d_ws;
    float*    X      = (float*)w;     w += (size_t)N * 64 * 4;
    float*    AGG    = (float*)w;     w += (size_t)N * 64 * 4;
    float*    SCORE  = (float*)w;     w += (size_t)N * 4;
    unsigned* KEY    = (unsigned*)w;  w += (size_t)N * 4;
    int*      MASK   = (int*)w;       w += (size_t)N * 4;
    int*      STARTS = (int*)w;       w += 256;
    float*    R      = (float*)w;     w += (size_t)GG * 128 * 4;

    const int nfeat = N * 64;
    const int fb    = (nfeat + 255) / 256;

    starts_kernel<<<1, 128, 0, stream>>>(batch, STARTS, N);
    fill_i_kernel<<<(N + 255) / 256, 256, 0, stream>>>(MASK, 1, N);
    fill_f_kernel<<<(GG * 128 + 255) / 256, 256, 0, stream>>>(R, 0.0f, GG * 128);
    copy_f4_kernel<<<(nfeat / 4 + 255) / 256, 256, 0, stream>>>(
        (float4*)X, (const float4*)x_in, nfeat / 4);

    for (int rd = 0; rd < 4; ++rd) {
        const int base = 4 + 5 * rd;
        const float* cW1 = (const float*)d_in[base + 0];
        const float* cb1 = (const float*)d_in[base + 1];
        const float* cW2 = (const float*)d_in[base + 2];
        const float* cb2 = (const float*)d_in[base + 3];
        const float* pw  = (const float*)d_in[base + 4];

        fill_f_kernel<<<fb, 256, 0, stream>>>(AGG, 0.0f, nfeat);
        scatter_agg_kernel<<<(E * 64) / 256, 256, 0, stream>>>(esrc, edst, X, AGG, E);
        gin_mlp_kernel<<<(N / 16 + 3) / 4, 128, 0, stream>>>(
            X, AGG, cW1, cb1, cW2, cb2, MASK, X, N);
        score_kernel<<<(N + 255) / 256, 256, 0, stream>>>(X, pw, MASK, SCORE, KEY, N);
        topk_kernel<<<GG, 256, 0, stream>>>(KEY, STARTS, MASK);
        gate_kernel<<<fb, 256, 0, stream>>>(X, SCORE, MASK, nfeat);
        readout_kernel<<<GG, 256, 0, stream>>>(X, MASK, STARTS, R);
    }

    final_mlp_kernel<<<1, 64, 0, stream>>>(
        R,
        (const float*)d_in[24], (const float*)d_in[25],
        (const float*)d_in[26], (const float*)d_in[27],
        (const float*)d_in[28], (const float*)d_in[29],
        (float*)d_out);
}